// TemporalLSTMGNN_50285477102146
// MI455X (gfx1250) — compile-verified
//
#include <hip/hip_runtime.h>
#include <hip/hip_bf16.h>
#include <math.h>

typedef __attribute__((ext_vector_type(2))) float v2f;
typedef __attribute__((ext_vector_type(8))) float v8f;

// ---------------- utility ----------------
__global__ void k_zero(float* p, long n) {
  long i = (long)blockIdx.x * blockDim.x + threadIdx.x;
  if (i < n) p[i] = 0.f;
}

// ---------------- edge weights / normalization ----------------
__global__ void k_edge_w(const float* __restrict__ ef, const float* __restrict__ W_edge,
                         const float* __restrict__ b_edge, const int* __restrict__ src,
                         float* __restrict__ w_e, float* __restrict__ deg, int E) {
  int e = blockIdx.x * blockDim.x + threadIdx.x;
  if (e >= E) return;
  float w = b_edge[0];
#pragma unroll
  for (int f = 0; f < 4; ++f) w += ef[e * 4 + f] * W_edge[f];
  w = fmaxf(w, 0.f);
  w_e[e] = w;
  atomicAdd(&deg[src[e]], w);
}

__global__ void k_norm(const float* __restrict__ w_e, const float* __restrict__ deg,
                       const int* __restrict__ src, const int* __restrict__ dst,
                       float* __restrict__ norm, int E) {
  int e = blockIdx.x * blockDim.x + threadIdx.x;
  if (e >= E) return;
  float ds = deg[src[e]], dd = deg[dst[e]];
  float is = ds > 0.f ? rsqrtf(ds) : 0.f;
  float id = dd > 0.f ? rsqrtf(dd) : 0.f;
  norm[e] = -is * w_e[e] * id;
}

// ---------------- weight repack for flattened GEMMs ----------------
// Wx [4,5,16,32] -> WX2 [(k*16+f) , (g*32+o)]  (80 x 128)
__global__ void k_packWX(const float* __restrict__ Wx, float* __restrict__ W2) {
  int i = blockIdx.x * blockDim.x + threadIdx.x;
  if (i >= 4 * 5 * 16 * 32) return;
  int o = i & 31, f = (i >> 5) & 15, gk = i >> 9, k = gk % 5, g = gk / 5;
  W2[(k * 16 + f) * 128 + g * 32 + o] = Wx[i];
}
// Wh [4,5,32,32] -> WH2 [(k*32+f) , (g*32+o)]  (160 x 128)
__global__ void k_packWH(const float* __restrict__ Wh, float* __restrict__ W2) {
  int i = blockIdx.x * blockDim.x + threadIdx.x;
  if (i >= 4 * 5 * 32 * 32) return;
  int o = i & 31, f = (i >> 5) & 31, gk = i >> 10, k = gk % 5, g = gk / 5;
  W2[(k * 32 + f) * 128 + g * 32 + o] = Wh[i];
}

// ---------------- per-timestep front end ----------------
// x = X_t @ W_feat^T + b_feat into Cheb slot 0 of TxX [N,80]; zero slots 1..4
__global__ void k_featlin(const float* __restrict__ Xt, const float* __restrict__ W_feat,
                          const float* __restrict__ b_feat, float* __restrict__ TxX, int N) {
  long t = (long)blockIdx.x * blockDim.x + threadIdx.x;
  if (t >= (long)N * 80) return;
  int n = (int)(t / 80), j = (int)(t % 80);
  float v = 0.f;
  if (j < 16) {
    v = b_feat[j];
#pragma unroll
    for (int i = 0; i < 16; ++i) v += Xt[n * 16 + i] * W_feat[j * 16 + i];
  }
  TxX[t] = v;
}

// copy H into Cheb slot 0 of TxH [N,160]; zero slots 1..4
__global__ void k_copyH(const float* __restrict__ H, float* __restrict__ TxH, int N) {
  long t = (long)blockIdx.x * blockDim.x + threadIdx.x;
  if (t >= (long)N * 160) return;
  int n = (int)(t / 160), j = (int)(t % 160);
  TxH[t] = (j < 32) ? H[n * 32 + j] : 0.f;
}

// sparse Laplacian apply: Tb[dst, offout+f] += norm[e] * Tb[src, offin+f]
__global__ void k_lap(float* __restrict__ Tb, int stride, int offin, int offout, int logF,
                      const int* __restrict__ src, const int* __restrict__ dst,
                      const float* __restrict__ norm, int E) {
  long t = (long)blockIdx.x * blockDim.x + threadIdx.x;
  if (t >= ((long)E << logF)) return;
  int e = (int)(t >> logF);
  int f = (int)(t & ((1 << logF) - 1));
  float v = norm[e] * Tb[(long)src[e] * stride + offin + f];
  atomicAdd(&Tb[(long)dst[e] * stride + offout + f], v);
}

// Chebyshev finalize: T_k = 2*lap(T_{k-1}) - T_{k-2}
__global__ void k_chebfin(float* __restrict__ Tb, int stride, int offk, int offkm2, int F, int N) {
  long t = (long)blockIdx.x * blockDim.x + threadIdx.x;
  if (t >= (long)N * F) return;
  int n = (int)(t / F), f = (int)(t % F);
  float* row = Tb + (long)n * stride;
  row[offk + f] = 2.f * row[offk + f] - row[offkm2 + f];
}

// ---------------- gate GEMM with WMMA (fp32) ----------------
// G[N,128] = TxX[N,80] @ WX2[80,128] + TxH[N,160] @ WH2[160,128]
// One wave computes one 16x16 output tile. 8 waves/block cover the 128 gate cols.
// A layout (32-bit 16x4): lanes 0-15 hold M rows; VGPR0 = {K0 | K2}, VGPR1 = {K1 | K3}
// C/D layout: VGPR i -> row m0+i (lanes 0-15) / m0+i+8 (lanes 16-31), col = lane&15
__global__ __launch_bounds__(256) void k_gates_gemm(
    const float* __restrict__ TxX, const float* __restrict__ TxH,
    const float* __restrict__ WX2, const float* __restrict__ WH2,
    float* __restrict__ G, int N) {
  const int wave = threadIdx.x >> 5;
  const int lane = threadIdx.x & 31;
  const int half = lane >> 4;   // 0: K pair {0,1}; 1: K pair {2,3}
  const int l16  = lane & 15;
  const int m0   = blockIdx.x * 16;
  int rowA = m0 + l16;
  if (rowA >= N) rowA = N - 1;  // clamp loads; stores are guarded
  const int col = (wave << 4) + l16;

  v8f acc = {};
  const float* ax = TxX + (long)rowA * 80;
#pragma unroll
  for (int kk = 0; kk < 80; kk += 4) {
    v2f a, b;
    a.x = ax[kk + half * 2 + 0];
    a.y = ax[kk + half * 2 + 1];
    b.x = WX2[(kk + half * 2 + 0) * 128 + col];
    b.y = WX2[(kk + half * 2 + 1) * 128 + col];
    acc = __builtin_amdgcn_wmma_f32_16x16x4_f32(false, a, false, b, (short)0, acc, false, false);
  }
  const float* ah = TxH + (long)rowA * 160;
#pragma unroll
  for (int kk = 0; kk < 160; kk += 4) {
    v2f a, b;
    a.x = ah[kk + half * 2 + 0];
    a.y = ah[kk + half * 2 + 1];
    b.x = WH2[(kk + half * 2 + 0) * 128 + col];
    b.y = WH2[(kk + half * 2 + 1) * 128 + col];
    acc = __builtin_amdgcn_wmma_f32_16x16x4_f32(false, a, false, b, (short)0, acc, false, false);
  }
#pragma unroll
  for (int i = 0; i < 8; ++i) {
    int row = m0 + i + half * 8;
    if (row < N) G[(long)row * 128 + col] = acc[i];
  }
}

// ---------------- LSTM pointwise ----------------
__global__ void k_lstm(const float* __restrict__ G, const float* __restrict__ w_peep,
                       const float* __restrict__ b_g, float* __restrict__ H,
                       float* __restrict__ C, int N) {
  long t = (long)blockIdx.x * blockDim.x + threadIdx.x;
  if (t >= (long)N * 32) return;
  int n = (int)(t >> 5), o = (int)(t & 31);
  const float* g = G + (long)n * 128;
  float c = C[t];
  float I  = 1.f / (1.f + __expf(-(g[o]       + w_peep[o]      * c  + b_g[o])));
  float Fg = 1.f / (1.f + __expf(-(g[32 + o]  + w_peep[32 + o] * c  + b_g[32 + o])));
  float Tt = tanhf(g[64 + o] + b_g[64 + o]);
  float cn = Fg * c + I * Tt;
  float O  = 1.f / (1.f + __expf(-(g[96 + o]  + w_peep[64 + o] * cn + b_g[96 + o])));
  C[t] = cn;
  H[t] = O * tanhf(cn);
}

// ---------------- output projection ----------------
__global__ void k_out(const float* __restrict__ H, const float* __restrict__ W_out,
                      const float* __restrict__ b_out, float* __restrict__ out, int N) {
  long t = (long)blockIdx.x * blockDim.x + threadIdx.x;
  if (t >= (long)N * 12) return;
  int n = (int)(t / 12), p = (int)(t % 12);
  float v = b_out[p];
#pragma unroll
  for (int o = 0; o < 32; ++o) v += fmaxf(H[n * 32 + o], 0.f) * W_out[p * 32 + o];
  out[t] = v;
}

extern "C" void kernel_launch(void* const* d_in, const int* in_sizes, int n_in,
                              void* d_out, int out_size, void* d_ws, size_t ws_size,
                              hipStream_t stream) {
  const float* timesteps = (const float*)d_in[0];
  const int*   eidx      = (const int*)d_in[1];
  const float* efeat     = (const float*)d_in[2];
  const float* Wx        = (const float*)d_in[3];
  const float* Wh        = (const float*)d_in[4];
  const float* w_peep    = (const float*)d_in[5];
  const float* b_gates   = (const float*)d_in[6];
  const float* W_feat    = (const float*)d_in[7];
  const float* b_feat    = (const float*)d_in[8];
  const float* W_edge    = (const float*)d_in[9];
  const float* b_edge    = (const float*)d_in[10];
  const float* W_out     = (const float*)d_in[11];
  const float* b_out     = (const float*)d_in[12];

  const int T = 12;
  const int N = in_sizes[0] / (T * 16);
  const int E = in_sizes[2] / 4;
  const int* src = eidx;
  const int* dst = eidx + E;

  float* ws  = (float*)d_ws;
  float* deg = ws;  ws += N;
  float* w_e = ws;  ws += E;
  float* nrm = ws;  ws += E;
  float* TxX = ws;  ws += (size_t)N * 80;
  float* TxH = ws;  ws += (size_t)N * 160;
  float* G   = ws;  ws += (size_t)N * 128;
  float* Hb  = ws;  ws += (size_t)N * 32;
  float* Cb  = ws;  ws += (size_t)N * 32;
  float* WX2 = ws;  ws += 80 * 128;
  float* WH2 = ws;  ws += 160 * 128;

  auto nb = [](long n) { return (unsigned)((n + 255) / 256); };

  // ---- one-time prep ----
  k_zero<<<nb(N), 256, 0, stream>>>(deg, N);
  k_zero<<<nb((long)N * 64), 256, 0, stream>>>(Hb, (long)N * 64);  // H and C contiguous
  k_edge_w<<<nb(E), 256, 0, stream>>>(efeat, W_edge, b_edge, src, w_e, deg, E);
  k_norm<<<nb(E), 256, 0, stream>>>(w_e, deg, src, dst, nrm, E);
  k_packWX<<<nb(10240), 256, 0, stream>>>(Wx, WX2);
  k_packWH<<<nb(20480), 256, 0, stream>>>(Wh, WH2);

  // ---- recurrent loop ----
  for (int t = 0; t < T; ++t) {
    const float* Xt = timesteps + (size_t)t * N * 16;
    k_featlin<<<nb((long)N * 80), 256, 0, stream>>>(Xt, W_feat, b_feat, TxX, N);
    k_copyH<<<nb((long)N * 160), 256, 0, stream>>>(Hb, TxH, N);
    for (int k = 1; k < 5; ++k) {
      k_lap<<<nb((long)E * 16), 256, 0, stream>>>(TxX, 80, (k - 1) * 16, k * 16, 4, src, dst, nrm, E);
      k_lap<<<nb((long)E * 32), 256, 0, stream>>>(TxH, 160, (k - 1) * 32, k * 32, 5, src, dst, nrm, E);
      if (k >= 2) {
        k_chebfin<<<nb((long)N * 16), 256, 0, stream>>>(TxX, 80, k * 16, (k - 2) * 16, 16, N);
        k_chebfin<<<nb((long)N * 32), 256, 0, stream>>>(TxH, 160, k * 32, (k - 2) * 32, 32, N);
      }
    }
    k_gates_gemm<<<(N + 15) / 16, 256, 0, stream>>>(TxX, TxH, WX2, WH2, G, N);
    k_lstm<<<nb((long)N * 32), 256, 0, stream>>>(G, w_peep, b_gates, Hb, Cb, N);
  }
  k_out<<<nb((long)N * 12), 256, 0, stream>>>(Hb, W_out, b_out, (float*)d_out, N);
}